// GCN_15960098472701
// MI455X (gfx1250) — compile-verified
//
#include <hip/hip_runtime.h>

#define N_NODES 100000
#define N_EDGES 1600000
#define IN_F    24
#define HID     128
#define OUT_F   12
#define NTILES  (N_NODES / 16)   // 6250, N divisible by 16

typedef __attribute__((ext_vector_type(2))) float v2f;
typedef __attribute__((ext_vector_type(8))) float v8f;

__device__ __forceinline__ v8f wmma_f32(v2f a, v2f b, v8f c) {
    // D = A(16x4) * B(4x16) + C(16x16), full fp32
    return __builtin_amdgcn_wmma_f32_16x16x4_f32(
        /*neg_a=*/false, a, /*neg_b=*/false, b,
        /*c_mod=*/(short)0, c, /*reuse_a=*/false, /*reuse_b=*/false);
}

// ---------------- utility: zero a float region ----------------
__global__ void k_zero(float* __restrict__ p, int n) {
    int i = blockIdx.x * blockDim.x + threadIdx.x;
    if (i < n) p[i] = 0.0f;
}

// ---------------- degrees via f32 atomics ----------------
__global__ void k_degree(const int* __restrict__ src, const int* __restrict__ dst,
                         float* __restrict__ dout, float* __restrict__ din) {
    int e = blockIdx.x * blockDim.x + threadIdx.x;
    if (e < N_EDGES) {
        atomicAdd(&dout[src[e]], 1.0f);
        atomicAdd(&din[dst[e]], 1.0f);
    }
}

// in-place: deg -> rsqrt(max(deg,1))
__global__ void k_scale(float* __restrict__ dout, float* __restrict__ din) {
    int i = blockIdx.x * blockDim.x + threadIdx.x;
    if (i < N_NODES) {
        dout[i] = rsqrtf(fmaxf(dout[i], 1.0f));
        din[i]  = rsqrtf(fmaxf(din[i], 1.0f));
    }
}

// ---------------- embed: h = X @ W_emb + b  (N x 24 x 128) ----------------
__global__ __launch_bounds__(256) void k_embed(const float* __restrict__ X,
                                               const float* __restrict__ W,
                                               const float* __restrict__ b,
                                               float* __restrict__ h) {
    __shared__ float Atile[16 * IN_F];
    const int tile = blockIdx.x;
    const int tid  = threadIdx.x;
    for (int i = tid; i < 16 * IN_F; i += 256) Atile[i] = X[tile * 16 * IN_F + i];
    __syncthreads();

    const int lane = tid & 31, wave = tid >> 5;
    const int m = lane & 15, half = lane >> 4;
    const int n = wave * 16 + m;          // global output column

    v8f acc = {};
    for (int kk = 0; kk < IN_F; kk += 4) {
        const int k0 = kk + 2 * half;
        v2f a = *(const v2f*)&Atile[m * IN_F + k0];
        v2f bb;
        bb.x = W[k0 * HID + n];
        bb.y = W[(k0 + 1) * HID + n];
        acc = wmma_f32(a, bb, acc);
    }
    const float bias = b[n];
#pragma unroll
    for (int r = 0; r < 8; ++r) {
        const int row = tile * 16 + r + 8 * half;
        h[row * HID + n] = acc[r] + bias;
    }
}

// ------- scatter: agg[dst] += h[src] * (e_w * out_deg^-0.5[src]) -------
// one thread per (edge, 4-float chunk): E*32 threads, float4 gather + 4 f32 atomics
__global__ __launch_bounds__(256) void k_scatter(const float* __restrict__ h,
                                                 const int* __restrict__ src,
                                                 const int* __restrict__ dst,
                                                 const float* __restrict__ ew,
                                                 const float* __restrict__ scout,
                                                 float* __restrict__ agg) {
    const int idx = blockIdx.x * blockDim.x + threadIdx.x;   // < E*32 exactly
    const int e = idx >> 5;
    const int c = (idx & 31) * 4;
    const int s = src[e];
    const int d = dst[e];
    const float w = ew[e] * scout[s];
    const float4 v = *(const float4*)&h[s * HID + c];
    float* o = &agg[d * HID + c];
    atomicAdd(o + 0, v.x * w);
    atomicAdd(o + 1, v.y * w);
    atomicAdd(o + 2, v.z * w);
    atomicAdd(o + 3, v.w * w);
}

// ------- combine: h = relu(h@Wself + (agg@W + b) * in_deg^-0.5), agg tile zeroed -------
__global__ __launch_bounds__(256) void k_combine(float* __restrict__ h,
                                                 float* __restrict__ agg,
                                                 const float* __restrict__ Wself,
                                                 const float* __restrict__ W,
                                                 const float* __restrict__ bias,
                                                 const float* __restrict__ scin) {
    __shared__ float Ah[16 * HID];
    __shared__ float Ag[16 * HID];
    const int tile = blockIdx.x, tid = threadIdx.x;
    const int base = tile * 16 * HID;
    for (int i = tid; i < 512; i += 256) {          // 16*128 floats as float4
        float4 vh = *(const float4*)&h[base + i * 4];
        float4 vg = *(const float4*)&agg[base + i * 4];
        *(float4*)&Ah[i * 4] = vh;
        *(float4*)&Ag[i * 4] = vg;
        *(float4*)&agg[base + i * 4] = make_float4(0.f, 0.f, 0.f, 0.f);  // reset for next layer
    }
    __syncthreads();

    const int lane = tid & 31, wave = tid >> 5;
    const int m = lane & 15, half = lane >> 4;
    const int n = wave * 16 + m;

    v8f accS = {}, accA = {};
    for (int kk = 0; kk < HID; kk += 4) {
        const int k0 = kk + 2 * half;
        v2f aS = *(const v2f*)&Ah[m * HID + k0];
        v2f aA = *(const v2f*)&Ag[m * HID + k0];
        v2f bS, bA;
        bS.x = Wself[k0 * HID + n];
        bS.y = Wself[(k0 + 1) * HID + n];
        bA.x = W[k0 * HID + n];
        bA.y = W[(k0 + 1) * HID + n];
        accS = wmma_f32(aS, bS, accS);
        accA = wmma_f32(aA, bA, accA);
    }
    const float bn = bias[n];
#pragma unroll
    for (int r = 0; r < 8; ++r) {
        const int row = tile * 16 + r + 8 * half;
        const float v = accS[r] + (accA[r] + bn) * scin[row];
        h[row * HID + n] = fmaxf(v, 0.0f);
    }
}

// ---------------- fc: out = h @ W_fc + b  (N x 128 x 12) ----------------
__global__ __launch_bounds__(256) void k_fc(const float* __restrict__ h,
                                            const float* __restrict__ Wfc,
                                            const float* __restrict__ bfc,
                                            float* __restrict__ out) {
    const int wave = threadIdx.x >> 5, lane = threadIdx.x & 31;
    const int tile = blockIdx.x * 8 + wave;   // one 16-row tile per wave
    if (tile >= NTILES) return;               // wave-uniform guard
    const int m = lane & 15, half = lane >> 4;
    const int n = m;

    v8f acc = {};
    for (int kk = 0; kk < HID; kk += 4) {
        const int k0 = kk + 2 * half;
        v2f a = *(const v2f*)&h[(tile * 16 + m) * HID + k0];
        v2f b;
        b.x = (n < OUT_F) ? Wfc[k0 * OUT_F + n] : 0.0f;
        b.y = (n < OUT_F) ? Wfc[(k0 + 1) * OUT_F + n] : 0.0f;
        acc = wmma_f32(a, b, acc);
    }
    if (n < OUT_F) {
        const float bn = bfc[n];
#pragma unroll
        for (int r = 0; r < 8; ++r) {
            const int row = tile * 16 + r + 8 * half;
            out[row * OUT_F + n] = acc[r] + bn;
        }
    }
}

extern "C" void kernel_launch(void* const* d_in, const int* in_sizes, int n_in,
                              void* d_out, int out_size, void* d_ws, size_t ws_size,
                              hipStream_t stream) {
    const float* X      = (const float*)d_in[0];
    const int*   src    = (const int*)d_in[1];
    const int*   dst    = (const int*)d_in[2];
    const float* ew     = (const float*)d_in[3];
    // d_in[4], d_in[5]: snorm_n / snorm_e unused by reference
    const float* W_emb  = (const float*)d_in[6];
    const float* b_emb  = (const float*)d_in[7];
    const float* W_s1   = (const float*)d_in[8];
    const float* W1     = (const float*)d_in[9];
    const float* b1     = (const float*)d_in[10];
    const float* W_s2   = (const float*)d_in[11];
    const float* W2     = (const float*)d_in[12];
    const float* b2     = (const float*)d_in[13];
    const float* W_fc   = (const float*)d_in[14];
    const float* b_fc   = (const float*)d_in[15];
    float* out = (float*)d_out;

    // workspace layout: h[N*128] | agg[N*128] | sc_out[N] | sc_in[N]
    float* h     = (float*)d_ws;
    float* agg   = h + (size_t)N_NODES * HID;
    float* scout = agg + (size_t)N_NODES * HID;
    float* scin  = scout + N_NODES;

    // zero agg + both degree arrays (contiguous: N*(128+2) floats) every call
    const int nz = N_NODES * (HID + 2);
    k_zero<<<(nz + 255) / 256, 256, 0, stream>>>(agg, nz);

    k_degree<<<(N_EDGES + 255) / 256, 256, 0, stream>>>(src, dst, scout, scin);
    k_scale<<<(N_NODES + 255) / 256, 256, 0, stream>>>(scout, scin);

    k_embed<<<NTILES, 256, 0, stream>>>(X, W_emb, b_emb, h);

    k_scatter<<<(N_EDGES * 32) / 256, 256, 0, stream>>>(h, src, dst, ew, scout, agg);
    k_combine<<<NTILES, 256, 0, stream>>>(h, agg, W_s1, W1, b1, scin);

    k_scatter<<<(N_EDGES * 32) / 256, 256, 0, stream>>>(h, src, dst, ew, scout, agg);
    k_combine<<<NTILES, 256, 0, stream>>>(h, agg, W_s2, W2, b2, scin);

    k_fc<<<(NTILES + 7) / 8, 256, 0, stream>>>(h, W_fc, b_fc, out);
}